// TestSelect_8512625180853
// MI455X (gfx1250) — compile-verified
//
#include <hip/hip_runtime.h>
#include <hip/hip_bf16.h>
#include <stdint.h>

// ---------------------------------------------------------------------------
// FCOS post-processing for MI455X (gfx1250).
// Memory-bound select/top-k: stream cls (131MB) once from HBM (pass 1 fills
// the 192MB L2), re-read from L2 in the compaction pass. Async global->LDS
// (ASYNCcnt) double-buffers the streaming read; histogram-based threshold +
// per-image bitonic sort gives an exact, deterministic top-1000.
// Grid is organized as one (image,class) row per blockIdx.y so the hot loops
// contain no integer division and centerness loads vectorize to b128.
// ---------------------------------------------------------------------------

#define TOPK        1000
#define NBINS       4096
#define CAP         4096                    // per-image candidate capacity (pow2, sort size)
#define KEEP_LOGIT  (-2.9444389791664403f)  // logit(0.05): sigmoid(x)>0.05 <=> x>this

#define K1_THREADS   256
#define STAGE_ELEMS  1024                   // 256 threads * 4 floats = 4KB per stage
#define K1_STAGES    8
#define K1_TILE      (STAGE_ELEMS * K1_STAGES)

__device__ __forceinline__ float fast_sigmoid(float x) {
  // v_exp_f32 + v_rcp_f32 instead of the IEEE divide expansion; scores are
  // only compared/binned, ~1ulp reciprocal is plenty (and used identically in
  // both passes, so the histogram threshold stays consistent).
  return __builtin_amdgcn_rcpf(1.0f + __expf(-x));
}

__device__ __forceinline__ uint32_t lds_lo32(const void* p) {
  // Generic pointers into LDS: flat aperture keeps the wave-relative LDS byte
  // offset in addr[31:0] (CDNA5 ISA 10.2) — exactly what the async
  // instruction's VDST VGPR wants.
  return (uint32_t)(uint64_t)(uintptr_t)p;
}

// ---------------------------------------------------------------------------
// Kernel 0: sigmoid(center) precompute + zero histograms/counters.
// ---------------------------------------------------------------------------
__global__ void init_kernel(const float* __restrict__ center,
                            float* __restrict__ ctrs,
                            unsigned* __restrict__ ghist,
                            unsigned* __restrict__ counts,
                            int* __restrict__ binB,
                            int N, int HW) {
  const int i = blockIdx.x * blockDim.x + threadIdx.x;
  if (i < N * HW)    ctrs[i]  = fast_sigmoid(center[i]);
  if (i < N * NBINS) ghist[i] = 0u;
  if (i < N)         { counts[i] = 0u; binB[i] = 0; }
}

// ---------------------------------------------------------------------------
// Kernel 1: one (n,c) row per blockIdx.y. Stream the row via async
// global->LDS (double buffered; each wave consumes only its own 512B slice so
// s_wait_asynccnt is the only sync needed), score, LDS histogram -> global.
// ---------------------------------------------------------------------------
__global__ __launch_bounds__(K1_THREADS)
void score_hist_kernel(const float* __restrict__ cls,
                       const float* __restrict__ ctrs,
                       unsigned* __restrict__ ghist,
                       int C, int HW) {
  __shared__ __align__(16) float buf[2][STAGE_ELEMS];
  __shared__ unsigned shist[NBINS];

  const int nc    = blockIdx.y;            // n*C + c
  const int n     = nc / C;                // per-block scalar math only
  const int tid   = threadIdx.x;
  const int tile0 = blockIdx.x * K1_TILE;  // offset within the HW row
  const float* crow = cls  + (size_t)nc * HW;
  const float* ctrb = ctrs + (size_t)n  * HW;

  for (int b = tid; b < NBINS; b += K1_THREADS) shist[b] = 0u;
  __syncthreads();

  auto issue = [&](int s) -> bool {
    const int estart = tile0 + s * STAGE_ELEMS;
    if (estart + STAGE_ELEMS > HW) return false;   // tail handled directly
    const uint32_t lds   = lds_lo32(&buf[s & 1][tid * 4]);
    const uint64_t gaddr = (uint64_t)(uintptr_t)(crow + estart + tid * 4);
    asm volatile("global_load_async_to_lds_b128 %0, %1, off"
                 :: "v"(lds), "v"(gaddr)
                 : "memory");
    return true;
  };

  (void)issue(0);
  for (int s = 0; s < K1_STAGES; ++s) {
    const int estart = tile0 + s * STAGE_ELEMS;
    if (estart >= HW) break;
    const bool full = (estart + STAGE_ELEMS <= HW);
    const bool nxt  = (s + 1 < K1_STAGES) ? issue(s + 1) : false;

    const int loc0 = estart + tid * 4;     // 4 consecutive spatial locations
    float x[4], ct[4];
    if (full) {
      if (nxt) asm volatile("s_wait_asynccnt 1" ::: "memory");
      else     asm volatile("s_wait_asynccnt 0" ::: "memory");
      const float4 v = *reinterpret_cast<const float4*>(&buf[s & 1][tid * 4]);
      x[0] = v.x; x[1] = v.y; x[2] = v.z; x[3] = v.w;
      const float4 cv = *reinterpret_cast<const float4*>(ctrb + loc0);
      ct[0] = cv.x; ct[1] = cv.y; ct[2] = cv.z; ct[3] = cv.w;
    } else {
#pragma unroll
      for (int j = 0; j < 4; ++j) {
        const bool ok = (loc0 + j < HW);
        x[j]  = ok ? crow[loc0 + j] : -1e30f;
        ct[j] = ok ? ctrb[loc0 + j] : 0.0f;
      }
    }
#pragma unroll
    for (int j = 0; j < 4; ++j) {
      if (x[j] > KEEP_LOGIT) {
        const float sc = fast_sigmoid(x[j]) * ct[j];
        int bin = (int)(sc * (float)NBINS);
        bin = bin < 0 ? 0 : (bin > NBINS - 1 ? NBINS - 1 : bin);
        atomicAdd(&shist[bin], 1u);
      }
    }
  }
  asm volatile("s_wait_asynccnt 0" ::: "memory");
  __syncthreads();
  for (int b = tid; b < NBINS; b += K1_THREADS) {
    const unsigned cnt = shist[b];
    if (cnt) atomicAdd(&ghist[n * NBINS + b], cnt);
  }
}

// ---------------------------------------------------------------------------
// Kernel 2: per-image threshold bin B: largest B with count(bins >= B) >= TOPK.
// ---------------------------------------------------------------------------
__global__ void threshold_kernel(const unsigned* __restrict__ ghist,
                                 int* __restrict__ binB) {
  __shared__ unsigned sbins[NBINS];
  __shared__ unsigned spart[256];
  const int n = blockIdx.x, tid = threadIdx.x;
  const int segw = NBINS / 256;

  for (int b = tid; b < NBINS; b += 256) sbins[b] = ghist[n * NBINS + b];
  __syncthreads();
  unsigned psum = 0;
  for (int b = tid * segw; b < (tid + 1) * segw; ++b) psum += sbins[b];
  spart[tid] = psum;
  __syncthreads();
  if (tid == 0) {
    unsigned acc = 0; int B = 0;
    for (int t = 255; t >= 0; --t) {
      if (acc + spart[t] >= (unsigned)TOPK) {
        for (int b = (t + 1) * segw - 1; b >= t * segw; --b) {
          acc += sbins[b];
          if (acc >= (unsigned)TOPK) { B = b; break; }
        }
        break;
      }
      acc += spart[t];
    }
    binB[n] = B;   // stays 0 if total candidates < TOPK -> take everything
  }
}

// ---------------------------------------------------------------------------
// Kernel 3: re-stream cls (hits the 192MB L2), compact candidates bin >= B.
// Identical score/bin arithmetic to kernel 1 so the threshold is consistent.
// ---------------------------------------------------------------------------
__global__ __launch_bounds__(256)
void compact_kernel(const float* __restrict__ cls,
                    const float* __restrict__ ctrs,
                    const int* __restrict__ binB,
                    unsigned* __restrict__ counts,
                    float* __restrict__ cscore,
                    int* __restrict__ cidx,
                    int C, int HW) {
  const int nc   = blockIdx.y;             // n*C + c
  const int n    = nc / C;
  const int c    = nc - n * C;
  const int loc0 = (blockIdx.x * blockDim.x + threadIdx.x) * 4;
  if (loc0 >= HW) return;
  const float* crow = cls  + (size_t)nc * HW;
  const float* ctrb = ctrs + (size_t)n  * HW;
  const int B = binB[n];

  __builtin_prefetch(crow + loc0 + 8192, 0, 1);   // global_prefetch_b8

  float x[4], ct[4];
  if (loc0 + 4 <= HW) {
    const float4 v = *reinterpret_cast<const float4*>(crow + loc0);
    x[0] = v.x; x[1] = v.y; x[2] = v.z; x[3] = v.w;
    const float4 cv = *reinterpret_cast<const float4*>(ctrb + loc0);
    ct[0] = cv.x; ct[1] = cv.y; ct[2] = cv.z; ct[3] = cv.w;
  } else {
#pragma unroll
    for (int j = 0; j < 4; ++j) {
      const bool ok = (loc0 + j < HW);
      x[j]  = ok ? crow[loc0 + j] : -1e30f;
      ct[j] = ok ? ctrb[loc0 + j] : 0.0f;
    }
  }
#pragma unroll
  for (int j = 0; j < 4; ++j) {
    if (x[j] > KEEP_LOGIT) {
      const float sc = fast_sigmoid(x[j]) * ct[j];
      int bin = (int)(sc * (float)NBINS);
      bin = bin < 0 ? 0 : (bin > NBINS - 1 ? NBINS - 1 : bin);
      if (bin >= B) {
        const unsigned p = atomicAdd(&counts[n], 1u);
        if (p < CAP) {
          cscore[(size_t)n * CAP + p] = sc;
          cidx  [(size_t)n * CAP + p] = (loc0 + j) * C + c;  // flat [HW,C] index
        }
      }
    }
  }
}

// ---------------------------------------------------------------------------
// Kernel 4: per-image bitonic sort (desc score, asc idx tie-break = top_k
// order; also canonicalizes atomic append order -> deterministic), emit.
// ---------------------------------------------------------------------------
__global__ __launch_bounds__(1024)
void sort_emit_kernel(const unsigned* __restrict__ counts,
                      const float* __restrict__ cscore,
                      const int* __restrict__ cidx,
                      const float* __restrict__ box,
                      const float* __restrict__ points,
                      const float* __restrict__ revise,
                      float* __restrict__ out,
                      int N, int C, int HW) {
  __shared__ float ss[CAP];
  __shared__ int   ii[CAP];
  const int n = blockIdx.x, tid = threadIdx.x, bdim = blockDim.x;

  unsigned K = counts[n]; if (K > CAP) K = CAP;
  for (int i = tid; i < CAP; i += bdim) {
    if (i < (int)K) { ss[i] = cscore[(size_t)n * CAP + i]; ii[i] = cidx[(size_t)n * CAP + i]; }
    else            { ss[i] = -1e30f;                      ii[i] = 0x7fffffff; }
  }
  __syncthreads();

  for (unsigned k = 2; k <= (unsigned)CAP; k <<= 1) {
    for (unsigned j = k >> 1; j > 0; j >>= 1) {
      for (unsigned i = tid; i < (unsigned)CAP; i += bdim) {
        const unsigned ixj = i ^ j;
        if (ixj > i) {
          const float s1 = ss[i], s2 = ss[ixj];
          const int   i1 = ii[i], i2 = ii[ixj];
          const bool b21 = (s2 > s1) || ((s2 == s1) && (i2 < i1)); // [ixj] better
          const bool b12 = (s1 > s2) || ((s1 == s2) && (i1 < i2)); // [i] better
          const bool sw  = ((i & k) == 0) ? b21 : b12;             // desc overall
          if (sw) { ss[i] = s2; ss[ixj] = s1; ii[i] = i2; ii[ixj] = i1; }
        }
      }
      __syncthreads();
    }
  }

  float* dets = out;                               // [N,TOPK,4]
  float* ocls = out + (size_t)N * TOPK * 4;        // [N,TOPK]
  float* oscr = ocls + (size_t)N * TOPK;           // [N,TOPK]
  for (int t = tid; t < TOPK; t += bdim) {
    const float s  = ss[t];
    const int   id = ii[t];
    const bool valid = s > -5e29f;                 // matches top_vals > NEG_INF*0.5
    float d0 = 0.f, d1 = 0.f, d2 = 0.f, d3 = 0.f, cf = 0.f, sf = 0.f;
    if (valid) {
      const int loc = id / C;
      const int c1  = id - loc * C + 1;
      const float px = points[2 * loc]     + revise[((size_t)n * HW + loc) * 2];
      const float py = points[2 * loc + 1] + revise[((size_t)n * HW + loc) * 2 + 1];
      const float* bb = box + (size_t)n * 4 * HW;  // [4,HW] within image
      const float b0 = bb[loc], b1 = bb[HW + loc], b2 = bb[2 * HW + loc], b3 = bb[3 * HW + loc];
      d0 = px - b0; d1 = py - b1; d2 = px + b2; d3 = py + b3;
      cf = (float)c1; sf = s;
    }
    const size_t o = (size_t)n * TOPK + t;
    dets[o * 4 + 0] = d0; dets[o * 4 + 1] = d1;
    dets[o * 4 + 2] = d2; dets[o * 4 + 3] = d3;
    ocls[o] = cf; oscr[o] = sf;
  }
}

// ---------------------------------------------------------------------------
extern "C" void kernel_launch(void* const* d_in, const int* in_sizes, int n_in,
                              void* d_out, int out_size, void* d_ws, size_t ws_size,
                              hipStream_t stream) {
  (void)n_in; (void)out_size; (void)ws_size;
  const float* box    = (const float*)d_in[0];
  const float* cls    = (const float*)d_in[1];
  const float* center = (const float*)d_in[2];
  const float* points = (const float*)d_in[3];
  const float* revise = (const float*)d_in[4];

  const int HW  = in_sizes[3] / 2;            // points: [HW,2]
  const int N   = in_sizes[2] / HW;           // center: [N,1,H,W]
  const int C   = in_sizes[1] / (N * HW);     // cls: [N,C,H,W]

  // Workspace carve-up (~2.9 MB for N=16): all chunks 256B aligned.
  char* w = (char*)d_ws;
  float*    ctrs   = (float*)w;    w += (size_t)N * HW * sizeof(float);
  unsigned* ghist  = (unsigned*)w; w += (size_t)N * NBINS * sizeof(unsigned);
  unsigned* counts = (unsigned*)w; w += 256;
  int*      binB   = (int*)w;      w += 256;
  float*    cscore = (float*)w;    w += (size_t)N * CAP * sizeof(float);
  int*      cidx   = (int*)w;

  {
    int total = N * HW; if (N * NBINS > total) total = N * NBINS;
    init_kernel<<<(total + 255) / 256, 256, 0, stream>>>(center, ctrs, ghist, counts, binB, N, HW);
  }
  {
    dim3 grid((HW + K1_TILE - 1) / K1_TILE, N * C);   // one (n,c) row per y
    score_hist_kernel<<<grid, K1_THREADS, 0, stream>>>(cls, ctrs, ghist, C, HW);
  }
  threshold_kernel<<<N, 256, 0, stream>>>(ghist, binB);
  {
    dim3 grid((HW + 1023) / 1024, N * C);             // one (n,c) row per y
    compact_kernel<<<grid, 256, 0, stream>>>(cls, ctrs, binB, counts, cscore, cidx, C, HW);
  }
  sort_emit_kernel<<<N, 1024, 0, stream>>>(counts, cscore, cidx, box, points, revise,
                                           (float*)d_out, N, C, HW);
}